// Profile_62517543961105
// MI455X (gfx1250) — compile-verified
//
#include <hip/hip_runtime.h>
#include <hip/hip_bf16.h>
#include <math.h>

#define B_   4096
#define D_   1024
#define K_   5
#define P_   1323
#define LOG_2PI 1.8378770664093453f

typedef __attribute__((ext_vector_type(2))) float v2f;
typedef __attribute__((ext_vector_type(8))) float v8f;

// ---------------------------------------------------------------------------
// Kernel 0: pack [W_mix | W_mean | W_scale] into zero-padded Wpad[D][64].
//   cols 0..15  <- W_mix  (5 valid)
//   cols 16..31 <- W_mean (12 valid)
//   cols 32..63 <- W_scale (30 valid)
// Lets the WMMA GEMM load B-fragments unconditionally (no divergence).
// ---------------------------------------------------------------------------
__global__ __launch_bounds__(256)
void pack_weights_kernel(const float* __restrict__ Wmix,
                         const float* __restrict__ Wmean,
                         const float* __restrict__ Wscale,
                         float* __restrict__ Wpad)
{
    int i = blockIdx.x * blockDim.x + threadIdx.x;   // i < D*64
    if (i >= D_ * 64) return;
    int k = i >> 6, n = i & 63;
    float v = 0.0f;
    if (n < 16)      { if (n < 5)       v = Wmix[(size_t)k * 5 + n]; }
    else if (n < 32) { int c = n - 16;  if (c < 12) v = Wmean[(size_t)k * 12 + c]; }
    else             { int c = n - 32;  if (c < 30) v = Wscale[(size_t)k * 30 + c]; }
    Wpad[i] = v;
}

// ---------------------------------------------------------------------------
// Kernel 1: fused GEMM  raw[B,64] = rep[B,1024] x Wpad[1024,64]
// One wave computes one 16x16 C tile via 256 x V_WMMA_F32_16X16X4_F32.
// Branch-free inner loop: EXEC stays all-1s (WMMA requirement).
// ---------------------------------------------------------------------------
__global__ __launch_bounds__(256)
void gemm_wmma_kernel(const float* __restrict__ rep,
                      const float* __restrict__ Wpad,
                      float* __restrict__ raw)
{
    const int wave   = blockIdx.x * (blockDim.x >> 5) + (threadIdx.x >> 5);
    const int lane   = threadIdx.x & 31;
    const int n_tile = wave & 3;        // 4 N tiles of 16 cols
    const int m_tile = wave >> 2;       // 256 M tiles
    const int row    = lane & 15;       // A fragment: M row
    const int kb     = (lane >> 4) * 2; // K sub-offset within 4-wide K step

    const int    m0   = m_tile * 16;
    const float* arow = rep + (size_t)(m0 + row) * D_;
    const int    ncol = n_tile * 16 + (lane & 15);   // B fragment: N col (padded)

    v8f acc = {};
    for (int k = 0; k < D_; k += 4) {
        v2f a = *(const v2f*)(arow + k + kb);        // aligned b64 load
        v2f b;
        b.x = Wpad[(size_t)(k + kb)     * 64 + ncol];
        b.y = Wpad[(size_t)(k + kb + 1) * 64 + ncol];
        acc = __builtin_amdgcn_wmma_f32_16x16x4_f32(
                  false, a, false, b, (short)0, acc, false, false);
    }

    // C/D layout: lane l holds N = l&15; VGPR v holds M = v + 8*(l>>4)
    const int ccol = lane & 15;
    const int crow = (lane >> 4) * 8;
#pragma unroll
    for (int v = 0; v < 8; ++v) {
        raw[(size_t)(m0 + crow + v) * 64 + n_tile * 16 + ccol] = acc[v];
    }
}

// ---------------------------------------------------------------------------
// Kernel 2: per-(b,k) parameter prep + L output.
// params[b][k] = {c, 1/L00, L10, 1/L11, L20, L21, 1/L22, m0, m1, m2}
//   c = log w_k - logdet_k - 1.5*log(2*pi)
// ---------------------------------------------------------------------------
__device__ __forceinline__ float softplusf(float x) {
    return fmaxf(x, 0.0f) + log1pf(__expf(-fabsf(x)));
}

__global__ __launch_bounds__(256)
void params_kernel(const float* __restrict__ raw,
                   const float* __restrict__ bmix,
                   const float* __restrict__ bmean,
                   const float* __restrict__ bscale,
                   float* __restrict__ params,
                   float* __restrict__ Lout)
{
    int b = blockIdx.x * blockDim.x + threadIdx.x;
    if (b >= B_) return;
    const float* r = raw + (size_t)b * 64;

    float lg[K_];
    float mx = -1e30f;
#pragma unroll
    for (int k = 0; k < K_; ++k) { lg[k] = r[k] + bmix[k]; mx = fmaxf(mx, lg[k]); }
    float s = 0.0f;
#pragma unroll
    for (int k = 0; k < K_; ++k) s += __expf(lg[k] - mx);
    float logZ = mx + __logf(s);

#pragma unroll
    for (int k = 0; k < K_; ++k) {
        float m0 = 0.0f, m1 = 0.0f, m2 = 0.0f;
        if (k > 0) {
            int j = (k - 1) * 3;
            m0 = r[16 + j]     + bmean[j];
            m1 = r[16 + j + 1] + bmean[j + 1];
            m2 = r[16 + j + 2] + bmean[j + 2];
        }
        int sc = k * 6;
        float s0 = r[32 + sc]     + bscale[sc];
        float s1 = r[32 + sc + 1] + bscale[sc + 1];
        float s2 = r[32 + sc + 2] + bscale[sc + 2];
        float s3 = r[32 + sc + 3] + bscale[sc + 3];
        float s4 = r[32 + sc + 4] + bscale[sc + 4];
        float s5 = r[32 + sc + 5] + bscale[sc + 5];

        float L00 = softplusf(s0), L10 = s1, L11 = softplusf(s2);
        float L20 = s3, L21 = s4,  L22 = softplusf(s5);
        float logdet = __logf(L00) + __logf(L11) + __logf(L22);
        float c = (lg[k] - logZ) - logdet - 1.5f * LOG_2PI;

        float* q = params + ((size_t)b * K_ + k) * 10;
        q[0] = c;          q[1] = 1.0f / L00; q[2] = L10;
        q[3] = 1.0f / L11; q[4] = L20;        q[5] = L21;
        q[6] = 1.0f / L22; q[7] = m0;         q[8] = m1;  q[9] = m2;

        float* Lp = Lout + ((size_t)b * K_ + k) * 9;
        Lp[0] = L00; Lp[1] = 0.0f; Lp[2] = 0.0f;
        Lp[3] = L10; Lp[4] = L11;  Lp[5] = 0.0f;
        Lp[6] = L20; Lp[7] = L21;  Lp[8] = L22;
    }
}

// ---------------------------------------------------------------------------
// Kernel 3: heavy pointwise stage.  32x32 (p,b) tiles staged through LDS to
// turn the [P,B,3] -> [B,P] transpose into coalesced loads AND stores.
// profile[b][p] = sum_k exp(c_k - 0.5 * ||L_k^{-1}(x - mu_k)||^2)
// ---------------------------------------------------------------------------
#define BT 32
#define PT 32

__global__ __launch_bounds__(256)
void profile_kernel(const float* __restrict__ dxyz,
                    const float* __restrict__ params,
                    float* __restrict__ profile)
{
    __shared__ float sx[PT][BT * 3 + 1];  // stride 97 (odd) -> conflict-free
    __shared__ float sp[BT][K_ * 10];

    const int p0 = blockIdx.x * PT;
    const int b0 = blockIdx.y * BT;
    const int t  = threadIdx.x;

    // stage per-batch params (broadcast-read later)
    for (int i = t; i < BT * K_ * 10; i += 256) {
        int bl = i / (K_ * 10), rr = i % (K_ * 10);
        sp[bl][rr] = params[(size_t)(b0 + bl) * (K_ * 10) + rr];
    }
    // stage dxyz tile: row p -> 96 contiguous floats (b0..b0+31, xyz) — coalesced
    for (int i = t; i < PT * BT * 3; i += 256) {
        int pl = i / (BT * 3), rr = i % (BT * 3);
        int p  = p0 + pl;
        float v = 0.0f;
        if (p < P_) v = dxyz[((size_t)p * B_ + b0) * 3 + rr];
        sx[pl][rr] = v;
    }
    __syncthreads();

    const int pl = t & 31;   // lane -> p : coalesced profile stores
    const int bq = t >> 5;   // wave -> b
    const int p  = p0 + pl;

#pragma unroll
    for (int pass = 0; pass < 4; ++pass) {
        const int bl = pass * 8 + bq;
        const float x0 = sx[pl][bl * 3 + 0];
        const float x1 = sx[pl][bl * 3 + 1];
        const float x2 = sx[pl][bl * 3 + 2];
        float acc = 0.0f;
#pragma unroll
        for (int k = 0; k < K_; ++k) {
            const float* q = &sp[bl][k * 10];
            float d0 = x0 - q[7];
            float d1 = x1 - q[8];
            float d2 = x2 - q[9];
            float z0 = d0 * q[1];
            float z1 = (d1 - q[2] * z0) * q[3];
            float z2 = (d2 - q[4] * z0 - q[5] * z1) * q[6];
            float quad = fmaf(z0, z0, fmaf(z1, z1, z2 * z2));
            acc += __expf(fmaf(quad, -0.5f, q[0]));  // exp(logsumexp) == sum of exps
        }
        if (p < P_) profile[(size_t)(b0 + bl) * P_ + p] = acc;
    }
}

// ---------------------------------------------------------------------------
extern "C" void kernel_launch(void* const* d_in, const int* in_sizes, int n_in,
                              void* d_out, int out_size, void* d_ws, size_t ws_size,
                              hipStream_t stream) {
    const float* rep    = (const float*)d_in[0];  // [B,D]
    const float* dxyz   = (const float*)d_in[1];  // [P,B,3]
    const float* Wmix   = (const float*)d_in[2];  // [D,5]
    const float* bmix   = (const float*)d_in[3];  // [5]
    const float* Wmean  = (const float*)d_in[4];  // [D,12]
    const float* bmean  = (const float*)d_in[5];  // [12]
    const float* Wscale = (const float*)d_in[6];  // [D,30]
    const float* bscale = (const float*)d_in[7];  // [30]

    float* profile = (float*)d_out;                    // [B,P]
    float* Lout    = (float*)d_out + (size_t)B_ * P_;  // [B,K,3,3]

    char*  ws     = (char*)d_ws;
    float* raw    = (float*)ws;                                   // [B,64]  1 MB
    float* params = (float*)(ws + (size_t)B_ * 64 * sizeof(float));       // [B,5,10]
    float* Wpad   = (float*)(ws + (size_t)B_ * 64 * sizeof(float)
                                + (size_t)B_ * K_ * 10 * sizeof(float));  // [D,64]

    // 0) zero-padded weight pack: D*64 elements
    pack_weights_kernel<<<(D_ * 64 + 255) / 256, 256, 0, stream>>>(Wmix, Wmean,
                                                                   Wscale, Wpad);

    // 1) WMMA GEMM: 1024 waves = 128 blocks x 8 waves
    gemm_wmma_kernel<<<128, 256, 0, stream>>>(rep, Wpad, raw);

    // 2) parameter prep: one thread per batch element
    params_kernel<<<(B_ + 255) / 256, 256, 0, stream>>>(raw, bmix, bmean, bscale,
                                                        params, Lout);

    // 3) pointwise mixture evaluation over (P,B)
    dim3 grid((P_ + PT - 1) / PT, B_ / BT);
    profile_kernel<<<grid, 256, 0, stream>>>(dxyz, params, profile);
}